// SortNode2Pin_24764781429525
// MI455X (gfx1250) — compile-verified
//
#include <hip/hip_runtime.h>

// ---------------------------------------------------------------------------
// SortNode2Pin: per-node segment arg-min over CSR pin segments.
//   out[i] = flat_node2pin[argmin_{j in [starts[i],starts[i+1])}
//                          sorted_pin_map[flat_node2pin[j]]]   (0 if empty)
// Memory-bound (~80 MB moved, all inputs L2-resident on MI455X's 192MB L2).
// One node per thread; CSR offsets staged block-wide into LDS via the
// gfx1250 async global->LDS DMA path (ASYNCcnt / s_wait_asynccnt).
// ---------------------------------------------------------------------------

#define BLOCK 256

#if defined(__HIP_DEVICE_COMPILE__)
#if __has_builtin(__builtin_amdgcn_global_load_async_to_lds_b32)
#define HAVE_ASYNC_LDS 1
#endif
#if __has_builtin(__builtin_amdgcn_s_wait_asynccnt)
#define HAVE_WAIT_ASYNC_BUILTIN 1
#endif
#endif

typedef __attribute__((address_space(1))) int g1_int;
typedef __attribute__((address_space(3))) int l3_int;

__device__ __forceinline__ void wait_asynccnt0() {
#if defined(HAVE_WAIT_ASYNC_BUILTIN)
  __builtin_amdgcn_s_wait_asynccnt(0);
#elif defined(__HIP_DEVICE_COMPILE__)
  asm volatile("s_wait_asynccnt 0" ::: "memory");
#endif
}

__global__ __launch_bounds__(BLOCK) void SortNode2Pin_segmin_kernel(
    const int* __restrict__ starts,   // [num_nodes + 1]
    const int* __restrict__ n2p,      // [num_pins]
    const int* __restrict__ keymap,   // [num_pins]
    int* __restrict__ out,            // [num_nodes]
    int num_nodes) {
  __shared__ int s_starts[BLOCK + 1];

  const int t = threadIdx.x;
  const int base = blockIdx.x * BLOCK;

  // ---- Stage starts[base .. base+BLOCK] into LDS (async DMA on gfx1250) ----
  {
    int idx = base + t;
    if (idx > num_nodes) idx = num_nodes;  // starts has num_nodes+1 entries
#if defined(HAVE_ASYNC_LDS)
    __builtin_amdgcn_global_load_async_to_lds_b32(
        (g1_int*)(starts + idx), (l3_int*)&s_starts[t], 0, 0);
    if (t == 0) {
      int idx2 = base + BLOCK;
      if (idx2 > num_nodes) idx2 = num_nodes;
      __builtin_amdgcn_global_load_async_to_lds_b32(
          (g1_int*)(starts + idx2), (l3_int*)&s_starts[BLOCK], 0, 0);
    }
    wait_asynccnt0();  // drain this wave's async->LDS ops
#else
    s_starts[t] = starts[idx];
    if (t == 0) {
      int idx2 = base + BLOCK;
      if (idx2 > num_nodes) idx2 = num_nodes;
      s_starts[BLOCK] = starts[idx2];
    }
#endif
  }
  __syncthreads();  // other waves' staged offsets visible

  // ---- Per-node segment arg-min ----
  const int node = base + t;
  if (node < num_nodes) {
    const int s = s_starts[t];
    const int e = s_starts[t + 1];
    int best_key = 0x7FFFFFFF;
    int best_pin = 0;  // empty segment -> 0, matches reference zeros init
    for (int j = s; j < e; ++j) {
      const int pin = n2p[j];          // contiguous stream within segment
      const int key = keymap[pin];     // random gather, L2-resident (32 MB)
      const bool better = key < best_key;
      best_key = better ? key : best_key;
      best_pin = better ? pin : best_pin;
    }
    out[node] = best_pin;
  }
}

// ---------------------------------------------------------------------------

extern "C" void kernel_launch(void* const* d_in, const int* in_sizes, int n_in,
                              void* d_out, int out_size, void* d_ws,
                              size_t ws_size, hipStream_t stream) {
  const int* starts = (const int*)d_in[0];  // flat_node2pin_start
  const int* n2p    = (const int*)d_in[1];  // flat_node2pin
  const int* keymap = (const int*)d_in[2];  // sorted_pin_map
  int* out = (int*)d_out;                   // int32 output

  const int num_nodes = in_sizes[0] - 1;    // starts has num_nodes+1 entries
  const int grid = (num_nodes + BLOCK - 1) / BLOCK;

  SortNode2Pin_segmin_kernel<<<grid, BLOCK, 0, stream>>>(starts, n2p, keymap,
                                                         out, num_nodes);
}